// S3UnetRotate_32358283608573
// MI455X (gfx1250) — compile-verified
//
#include <hip/hip_runtime.h>
#include <hip/hip_bf16.h>
#include <math.h>

typedef float    v2f  __attribute__((ext_vector_type(2)));
typedef float    v8f  __attribute__((ext_vector_type(8)));

// ---------------------------------------------------------------------------
// Model constants
// ---------------------------------------------------------------------------
namespace {
constexpr int LN[5] = {40962, 10242, 2562, 642, 162};
constexpr int CH[6] = {2, 64, 128, 256, 512, 1024};
constexpr int N0 = 40962;
// d_in leaf indices (jax pytree flatten: dict keys sorted)
constexpr int I_D01 = 0, I_D02 = 1, I_D12 = 2;
constexpr int I_NEIGH0 = 3;           // 3..7
constexpr int I_TRI = 8;
constexpr int I_UPDOWN0 = 9;          // 9..12
constexpr int I_UPTOP0 = 13;          // 13..16
constexpr int I_MXYZ = 17;
constexpr int I_PARAM0 = 18;          // 3 unets x 82 leaves
constexpr int UNET_STRIDE = 82;
constexpr int I_R01 = 264, I_R02 = 265, I_R12 = 266, I_R20 = 267;
constexpr int I_X = 268;
}

__device__ __forceinline__ v8f wmma4(v2f a, v2f b, v8f c) {
  return __builtin_amdgcn_wmma_f32_16x16x4_f32(false, a, false, b,
                                               (short)0, c, false, false);
}

__device__ __forceinline__ v2f ld2(const float* p) {
  return *reinterpret_cast<const v2f*>(p);          // 8B-aligned by construction
}

// Shared epilogue: 16x64 tile, C/D layout: vgpr v -> row m0+v(+8), col = lane&15
__device__ __forceinline__ void store_tile(float* __restrict__ Y,
                                           const float* __restrict__ bias,
                                           int M, int N, int m0, int n0,
                                           int l15, int half,
                                           const v8f& a0, const v8f& a1,
                                           const v8f& a2, const v8f& a3) {
#pragma unroll
  for (int t = 0; t < 4; ++t) {
    const v8f& acc = (t == 0) ? a0 : (t == 1) ? a1 : (t == 2) ? a2 : a3;
    const int col = n0 + t * 16 + l15;
    const float bv = bias ? bias[col] : 0.f;
#pragma unroll
    for (int v = 0; v < 8; ++v) {
      int row = m0 + v + half * 8;
      if (row < M) Y[(size_t)row * N + col] = acc[v] + bv;
    }
  }
}

// ---------------------------------------------------------------------------
// Plain GEMM (no gather): Y[MxN] = X[MxK] @ W^T + b.  K%4==0, N%64==0.
// One wave -> 16x64 tile, 4 accumulators, straight-line inner loop.
// ---------------------------------------------------------------------------
__launch_bounds__(128)
__global__ void k_gemm_lin(const float* __restrict__ X, const float* __restrict__ W,
                           const float* __restrict__ bias, float* __restrict__ Y,
                           int M, int N, int K) {
  const int lane = threadIdx.x & 31, wave = threadIdx.x >> 5;
  const int tilesN = N >> 6;
  const int tilesM = (M + 15) >> 4;
  const int tile = blockIdx.x * 4 + wave;
  if (tile >= tilesM * tilesN) return;              // wave-uniform
  const int tm = tile / tilesN, tn = tile - tm * tilesN;
  const int m0 = tm << 4, n0 = tn << 6;
  const int l15 = lane & 15, half = lane >> 4;
  const int mrow = min(m0 + l15, M - 1);            // clamp: no guards in loop
  const float* xr = X + (size_t)mrow * K;
  const float* w0 = W + (size_t)(n0 + l15) * K;
  const float* w1 = w0 + (size_t)16 * K;
  const float* w2 = w0 + (size_t)32 * K;
  const float* w3 = w0 + (size_t)48 * K;
  v8f a0 = {}, a1 = {}, a2 = {}, a3 = {};
#pragma unroll 2
  for (int k0 = 0; k0 < K; k0 += 4) {
    const int ka = k0 + half * 2;                   // lanes 16-31: K rows +2
    v2f a  = ld2(xr + ka);
    v2f b0 = ld2(w0 + ka);
    v2f b1 = ld2(w1 + ka);
    v2f b2 = ld2(w2 + ka);
    v2f b3 = ld2(w3 + ka);
    a0 = wmma4(a, b0, a0);
    a1 = wmma4(a, b1, a1);
    a2 = wmma4(a, b2, a2);
    a3 = wmma4(a, b3, a3);
  }
  store_tile(Y, bias, M, N, m0, n0, l15, half, a0, a1, a2, a3);
}

// ---------------------------------------------------------------------------
// Gathered conv GEMM: A row m = concat_j X[gath[7m+j], 0:ic], ic%4==0, N%64==0.
// j-loop + c0-loop: no division, no guards.
// ---------------------------------------------------------------------------
__launch_bounds__(128)
__global__ void k_gemm_g4(const float* __restrict__ X, const int* __restrict__ gath,
                          int ic, const float* __restrict__ W,
                          const float* __restrict__ bias, float* __restrict__ Y,
                          int M, int N) {
  const int K = 7 * ic;
  const int lane = threadIdx.x & 31, wave = threadIdx.x >> 5;
  const int tilesN = N >> 6;
  const int tilesM = (M + 15) >> 4;
  const int tile = blockIdx.x * 4 + wave;
  if (tile >= tilesM * tilesN) return;
  const int tm = tile / tilesN, tn = tile - tm * tilesN;
  const int m0 = tm << 4, n0 = tn << 6;
  const int l15 = lane & 15, half = lane >> 4;
  const int mrow = min(m0 + l15, M - 1);
  const int* gv = gath + 7 * (size_t)mrow;
  const float* w0 = W + (size_t)(n0 + l15) * K;
  const float* w1 = w0 + (size_t)16 * K;
  const float* w2 = w0 + (size_t)32 * K;
  const float* w3 = w0 + (size_t)48 * K;
  v8f a0 = {}, a1 = {}, a2 = {}, a3 = {};
  for (int j = 0; j < 7; ++j) {
    const float* xr  = X + (size_t)gv[j] * ic;
    const float* wj0 = w0 + j * ic;
    const float* wj1 = w1 + j * ic;
    const float* wj2 = w2 + j * ic;
    const float* wj3 = w3 + j * ic;
#pragma unroll 2
    for (int c0 = 0; c0 < ic; c0 += 4) {
      const int ka = c0 + half * 2;
      v2f a  = ld2(xr + ka);
      v2f b0 = ld2(wj0 + ka);
      v2f b1 = ld2(wj1 + ka);
      v2f b2 = ld2(wj2 + ka);
      v2f b3 = ld2(wj3 + ka);
      a0 = wmma4(a, b0, a0);
      a1 = wmma4(a, b1, a1);
      a2 = wmma4(a, b2, a2);
      a3 = wmma4(a, b3, a3);
    }
  }
  store_tile(Y, bias, M, N, m0, n0, l15, half, a0, a1, a2, a3);
}

// ---------------------------------------------------------------------------
// Level-0 c1 special case: ic=2, K=14, N=64. Fully unrolled, 4 WMMA K-steps.
// K rows 14,15 are zeroed through the A fragment (so B padding is irrelevant).
// ---------------------------------------------------------------------------
__launch_bounds__(128)
__global__ void k_gemm_g2(const float* __restrict__ X, const int* __restrict__ gath,
                          const float* __restrict__ W, const float* __restrict__ bias,
                          float* __restrict__ Y, int M, int N) {
  constexpr int K = 14;
  const int lane = threadIdx.x & 31, wave = threadIdx.x >> 5;
  const int tilesN = N >> 6;                        // == 1
  const int tilesM = (M + 15) >> 4;
  const int tile = blockIdx.x * 4 + wave;
  if (tile >= tilesM * tilesN) return;
  const int tm = tile / tilesN, tn = tile - tm * tilesN;
  const int m0 = tm << 4, n0 = tn << 6;
  const int l15 = lane & 15, half = lane >> 4;
  const int mrow = min(m0 + l15, M - 1);
  const int* gv = gath + 7 * (size_t)mrow;
  const float* w0 = W + (size_t)(n0 + l15) * K;
  const float* w1 = w0 + (size_t)16 * K;
  const float* w2 = w0 + (size_t)32 * K;
  const float* w3 = w0 + (size_t)48 * K;
  v8f a0 = {}, a1 = {}, a2 = {}, a3 = {};
#pragma unroll
  for (int jj = 0; jj < 4; ++jj) {
    const int j = 2 * jj + half;                    // half 0 -> even j, half 1 -> odd j
    const float msk = (j < 7) ? 1.f : 0.f;          // only jj==3, half==1 invalid
    const int jc = (j < 7) ? j : 6;
    v2f a = ld2(X + (size_t)gv[jc] * 2);
    a.x *= msk; a.y *= msk;
    const int kb = 4 * jj + half * 2;
    const int kc = (kb + 1 < K) ? kb : 0;           // clamp address; value masked via A
    v2f b0 = ld2(w0 + kc);
    v2f b1 = ld2(w1 + kc);
    v2f b2 = ld2(w2 + kc);
    v2f b3 = ld2(w3 + kc);
    a0 = wmma4(a, b0, a0);
    a1 = wmma4(a, b1, a1);
    a2 = wmma4(a, b2, a2);
    a3 = wmma4(a, b3, a3);
  }
  store_tile(Y, bias, M, N, m0, n0, l15, half, a0, a1, a2, a3);
}

// ---------------------------------------------------------------------------
// 7-neighbor mean pooling
// ---------------------------------------------------------------------------
__global__ void k_pool7(const float* __restrict__ X, const int* __restrict__ nb,
                        float* __restrict__ Y, int Nout, int C) {
  int gid = blockIdx.x * blockDim.x + threadIdx.x;
  if (gid >= Nout * C) return;
  int v = gid / C, c = gid - v * C;
  float s = 0.f;
#pragma unroll
  for (int j = 0; j < 7; ++j) s += X[(size_t)nb[7 * v + j] * C + c];
  Y[gid] = s * (1.f / 7.f);
}

// ---------------------------------------------------------------------------
// BatchNorm stats: one block per channel -> st[2c]=mean, st[2c+1]=rsqrt(var+eps)
// ---------------------------------------------------------------------------
__launch_bounds__(256)
__global__ void k_bn_stats(const float* __restrict__ X, int Nr, int C,
                           float* __restrict__ st) {
  __shared__ float ss[256], sq[256];
  int c = blockIdx.x;
  float s = 0.f, q = 0.f;
  for (int r = threadIdx.x; r < Nr; r += 256) {
    float v = X[(size_t)r * C + c];
    s += v; q += v * v;
  }
  ss[threadIdx.x] = s; sq[threadIdx.x] = q;
  __syncthreads();
  for (int o = 128; o > 0; o >>= 1) {
    if (threadIdx.x < o) {
      ss[threadIdx.x] += ss[threadIdx.x + o];
      sq[threadIdx.x] += sq[threadIdx.x + o];
    }
    __syncthreads();
  }
  if (threadIdx.x == 0) {
    float m = ss[0] / (float)Nr;
    float var = sq[0] / (float)Nr - m * m;
    st[2 * c] = m;
    st[2 * c + 1] = rsqrtf(var + 1e-5f);
  }
}

__global__ void k_bn_lrelu(const float* __restrict__ X, const float* __restrict__ st,
                           const float* __restrict__ g, const float* __restrict__ be,
                           float* __restrict__ Y, int Nr, int C) {
  int gid = blockIdx.x * blockDim.x + threadIdx.x;
  if (gid >= Nr * C) return;
  int c = gid % C;
  float y = (X[gid] - st[2 * c]) * st[2 * c + 1] * g[c] + be[c];
  Y[gid] = (y >= 0.f) ? y : 0.2f * y;
}

// ---------------------------------------------------------------------------
// Upsample + concat: cat[Nf x 2oc] = [x1;x2 | skip]
// ---------------------------------------------------------------------------
__global__ void k_upcat(const float* __restrict__ yb, const int* __restrict__ top,
                        const int* __restrict__ down, const float* __restrict__ skip,
                        float* __restrict__ cat, int Np, int Nf, int oc) {
  int gid = blockIdx.x * blockDim.x + threadIdx.x;
  if (gid >= Nf * oc) return;
  int r = gid / oc, c = gid - r * oc;
  float v;
  if (r < Np) {
    v = yb[(size_t)top[r] * oc + c];
  } else {
    int r2 = r - Np;
    v = 0.5f * (yb[(size_t)down[2 * r2] * oc + c] +
                yb[(size_t)down[2 * r2 + 1] * oc + c]);
  }
  cat[(size_t)r * (2 * oc) + c] = v;
  cat[(size_t)r * (2 * oc) + oc + c] = skip[gid];
}

// ---------------------------------------------------------------------------
// Output head: P[r,o] = (H[r,:64] . W[o,:64] + b[o]) / 50
// ---------------------------------------------------------------------------
__global__ void k_out3(const float* __restrict__ H, const float* __restrict__ W,
                       const float* __restrict__ b, float* __restrict__ P, int Nr) {
  int gid = blockIdx.x * blockDim.x + threadIdx.x;
  if (gid >= Nr * 3) return;
  int r = gid / 3, o = gid - 3 * r;
  const float* h = H + (size_t)r * 64;
  const float* w = W + (size_t)o * 64;
  float s = 0.f;
#pragma unroll 8
  for (int k = 0; k < 64; ++k) s += h[k] * w[k];
  P[gid] = (s + b[o]) * 0.02f;
}

// ---------------------------------------------------------------------------
// phi assembly + final fusion
// ---------------------------------------------------------------------------
__device__ __forceinline__ void rot3(const float* __restrict__ R, const float* v,
                                     float* o) {
  o[0] = R[0] * v[0] + R[1] * v[1] + R[2] * v[2];
  o[1] = R[3] * v[0] + R[4] * v[1] + R[5] * v[2];
  o[2] = R[6] * v[0] + R[7] * v[1] + R[8] * v[2];
}

__global__ void k_phi(int mode, const int* __restrict__ setIdx, int n,
                      const float* __restrict__ p0, const float* __restrict__ p1,
                      const float* __restrict__ p2, const float* __restrict__ r01,
                      const float* __restrict__ r02, const float* __restrict__ r12,
                      float* __restrict__ phi) {
  int t = blockIdx.x * blockDim.x + threadIdx.x;
  if (t >= n) return;
  int i = setIdx[t];
  const float* P0 = p0 + 3 * (size_t)i;
  const float* P1 = p1 + 3 * (size_t)i;
  const float* P2 = p2 + 3 * (size_t)i;
  float a[3], b[3], o[3];
  if (mode == 0) {                       // d02: (p02 + p2)/2
    rot3(r02, P0, a);
    o[0] = 0.5f * (a[0] + P2[0]); o[1] = 0.5f * (a[1] + P2[1]); o[2] = 0.5f * (a[2] + P2[2]);
  } else if (mode == 1) {                // d12: (p12 + p2)/2
    rot3(r12, P1, a);
    o[0] = 0.5f * (a[0] + P2[0]); o[1] = 0.5f * (a[1] + P2[1]); o[2] = 0.5f * (a[2] + P2[2]);
  } else if (mode == 2) {                // d01: ((p01 + p1)/2) @ r12^T
    rot3(r01, P0, a);
    b[0] = 0.5f * (a[0] + P1[0]); b[1] = 0.5f * (a[1] + P1[1]); b[2] = 0.5f * (a[2] + P1[2]);
    rot3(r12, b, o);
  } else {                               // tri: (p12 + p2 + p02)/3
    rot3(r12, P1, a);
    rot3(r02, P0, b);
    const float t3 = 1.f / 3.f;
    o[0] = (a[0] + P2[0] + b[0]) * t3;
    o[1] = (a[1] + P2[1] + b[1]) * t3;
    o[2] = (a[2] + P2[2] + b[2]) * t3;
  }
  phi[3 * (size_t)i + 0] = o[0];
  phi[3 * (size_t)i + 1] = o[1];
  phi[3 * (size_t)i + 2] = o[2];
}

__global__ void k_final(const float* __restrict__ p0, const float* __restrict__ p1,
                        const float* __restrict__ p2, const float* __restrict__ phi,
                        const float* __restrict__ mxyz, const float* __restrict__ r01,
                        const float* __restrict__ r02, const float* __restrict__ r12,
                        const float* __restrict__ r20, float* __restrict__ out,
                        int Nr) {
  int i = blockIdx.x * blockDim.x + threadIdx.x;
  if (i >= Nr) return;
  const size_t S = (size_t)Nr * 3;
  const float* P0 = p0 + 3 * (size_t)i;
  const float* P1 = p1 + 3 * (size_t)i;
  const float* P2 = p2 + 3 * (size_t)i;
  float p01v[3], p12v[3], p02v[3], po[3];
  rot3(r01, P0, p01v);
  rot3(r12, P1, p12v);
  rot3(r02, P0, p02v);
  rot3(r20, phi + 3 * (size_t)i, po);
#pragma unroll
  for (int j = 0; j < 3; ++j) {
    out[S * 1 + 3 * (size_t)i + j] = po[j];
    out[S * 2 + 3 * (size_t)i + j] = p01v[j];
    out[S * 3 + 3 * (size_t)i + j] = P1[j];
    out[S * 4 + 3 * (size_t)i + j] = p12v[j];
    out[S * 5 + 3 * (size_t)i + j] = P2[j];
    out[S * 6 + 3 * (size_t)i + j] = p02v[j];
  }
  float a = po[0], bb = po[1], g = po[2];
  float ca = cosf(a), sa = sinf(a);
  float cb = cosf(bb), sb = sinf(bb);
  float cg = cosf(g), sg = sinf(g);
  float r1x = cg * cb, r1y = cg * sb * sa - sg * ca, r1z = sg * sa + cg * ca * sb;
  float r2x = cb * sg, r2y = cg * ca + sg * sb * sa, r2z = ca * sg * sb - cg * sa;
  float r3x = -sb,     r3y = cb * sa,                r3z = cb * ca;
  float mx = mxyz[3 * (size_t)i], my = mxyz[3 * (size_t)i + 1], mz = mxyz[3 * (size_t)i + 2];
  out[3 * (size_t)i + 0] = mx * r1x + my * r1y + mz * r1z;
  out[3 * (size_t)i + 1] = mx * r2x + my * r2y + mz * r2z;
  out[3 * (size_t)i + 2] = mx * r3x + my * r3y + mz * r3z;
}

// ---------------------------------------------------------------------------
// Host orchestration
// ---------------------------------------------------------------------------
static inline int ceil_div(int a, int b) { return (a + b - 1) / b; }

static inline void gemm(hipStream_t s, const float* X, const int* g, int ic,
                        const float* W, const float* b, float* Y,
                        int M, int N, int K) {
  int tiles = ((M + 15) / 16) * (N / 64);
  int blocks = (tiles + 3) / 4;
  if (!g) {
    k_gemm_lin<<<blocks, 128, 0, s>>>(X, W, b, Y, M, N, K);
  } else if (ic == 2) {
    k_gemm_g2<<<blocks, 128, 0, s>>>(X, g, W, b, Y, M, N);
  } else {
    k_gemm_g4<<<blocks, 128, 0, s>>>(X, g, ic, W, b, Y, M, N);
  }
}

extern "C" void kernel_launch(void* const* d_in, const int* in_sizes, int n_in,
                              void* d_out, int out_size, void* d_ws, size_t ws_size,
                              hipStream_t stream) {
  (void)in_sizes; (void)n_in; (void)out_size; (void)ws_size;

  const float* x    = (const float*)d_in[I_X];
  const float* mxyz = (const float*)d_in[I_MXYZ];
  const float* r01  = (const float*)d_in[I_R01];
  const float* r02  = (const float*)d_in[I_R02];
  const float* r12  = (const float*)d_in[I_R12];
  const float* r20  = (const float*)d_in[I_R20];
  const int* neigh[5];
  for (int l = 0; l < 5; ++l) neigh[l] = (const int*)d_in[I_NEIGH0 + l];
  const int* uptop[4];
  const int* updown[4];
  for (int l = 0; l < 4; ++l) {
    uptop[l]  = (const int*)d_in[I_UPTOP0 + l];
    updown[l] = (const int*)d_in[I_UPDOWN0 + l];
  }
  const int* d01 = (const int*)d_in[I_D01];
  const int* d02 = (const int*)d_in[I_D02];
  const int* d12 = (const int*)d_in[I_D12];
  const int* tri = (const int*)d_in[I_TRI];

  // workspace carve
  size_t off = 0;
  char* base = (char*)d_ws;
  auto carve = [&](size_t nfloats) -> float* {
    float* p = (float*)(base + off);
    off += ((nfloats * sizeof(float)) + 255) & ~(size_t)255;
    return p;
  };
  float* xsb[6];
  xsb[0] = nullptr;
  for (int i = 0; i < 5; ++i) xsb[i + 1] = carve((size_t)LN[i] * CH[i + 1]);
  float* pool = carve((size_t)LN[1] * CH[1]);
  float* ybuf = carve((size_t)LN[1] * 7 * CH[1]);
  float* cat  = carve((size_t)LN[0] * 2 * CH[1]);
  float* t1   = carve((size_t)LN[0] * CH[1]);
  float* t2   = carve((size_t)LN[0] * CH[1]);
  float* st   = carve(2 * 1024);
  float* pbuf[3];
  for (int u = 0; u < 3; ++u) pbuf[u] = carve((size_t)N0 * 3);
  float* phib = carve((size_t)N0 * 3);

  for (int u = 0; u < 3; ++u) {
    void* const* P = d_in + I_PARAM0 + u * UNET_STRIDE;
    // --- down path ---
    for (int i = 0; i < 5; ++i) {
      const int Ni = LN[i], ic = CH[i], oc = CH[i + 1];
      const float* src;
      if (i == 0) {
        src = x;
      } else {
        int n = Ni * ic;
        k_pool7<<<ceil_div(n, 256), 256, 0, stream>>>(xsb[i], neigh[i - 1], pool, Ni, ic);
        src = pool;
      }
      const float* be1 = (const float*)P[i * 8 + 0];
      const float* g1  = (const float*)P[i * 8 + 1];
      const float* be2 = (const float*)P[i * 8 + 2];
      const float* g2  = (const float*)P[i * 8 + 3];
      const float* c1W = (const float*)P[i * 8 + 4];
      const float* c1b = (const float*)P[i * 8 + 5];
      const float* c2W = (const float*)P[i * 8 + 6];
      const float* c2b = (const float*)P[i * 8 + 7];
      gemm(stream, src, neigh[i], ic, c1W, c1b, t1, Ni, oc, 7 * ic);
      k_bn_stats<<<oc, 256, 0, stream>>>(t1, Ni, oc, st);
      k_bn_lrelu<<<ceil_div(Ni * oc, 256), 256, 0, stream>>>(t1, st, g1, be1, t2, Ni, oc);
      gemm(stream, t2, neigh[i], oc, c2W, c2b, t1, Ni, oc, 7 * oc);
      k_bn_stats<<<oc, 256, 0, stream>>>(t1, Ni, oc, st);
      k_bn_lrelu<<<ceil_div(Ni * oc, 256), 256, 0, stream>>>(t1, st, g2, be2, xsb[i + 1], Ni, oc);
    }
    // --- up path ---
    const float* h = xsb[5];
    int hC = CH[5];
    for (int i = 0; i < 4; ++i) {
      const int l = 3 - i;
      const int Np = LN[l + 1], Nf = LN[l], oc = CH[l + 1];
      const float* be1 = (const float*)P[42 + i * 10 + 0];
      const float* g1  = (const float*)P[42 + i * 10 + 1];
      const float* be2 = (const float*)P[42 + i * 10 + 2];
      const float* g2  = (const float*)P[42 + i * 10 + 3];
      const float* c1W = (const float*)P[42 + i * 10 + 4];
      const float* c1b = (const float*)P[42 + i * 10 + 5];
      const float* c2W = (const float*)P[42 + i * 10 + 6];
      const float* c2b = (const float*)P[42 + i * 10 + 7];
      const float* upW = (const float*)P[42 + i * 10 + 8];
      const float* upb = (const float*)P[42 + i * 10 + 9];
      gemm(stream, h, nullptr, hC, upW, upb, ybuf, Np, 7 * oc, hC);
      k_upcat<<<ceil_div(Nf * oc, 256), 256, 0, stream>>>(ybuf, uptop[l], updown[l],
                                                          xsb[l + 1], cat, Np, Nf, oc);
      gemm(stream, cat, neigh[l], 2 * oc, c1W, c1b, t1, Nf, oc, 14 * oc);
      k_bn_stats<<<oc, 256, 0, stream>>>(t1, Nf, oc, st);
      k_bn_lrelu<<<ceil_div(Nf * oc, 256), 256, 0, stream>>>(t1, st, g1, be1, t2, Nf, oc);
      gemm(stream, t2, neigh[l], oc, c2W, c2b, t1, Nf, oc, 7 * oc);
      k_bn_stats<<<oc, 256, 0, stream>>>(t1, Nf, oc, st);
      k_bn_lrelu<<<ceil_div(Nf * oc, 256), 256, 0, stream>>>(t1, st, g2, be2, xsb[l + 1], Nf, oc);
      h = xsb[l + 1];
      hC = oc;
    }
    // --- output head (/50 fused) ---
    const float* oW = (const float*)P[40];
    const float* ob = (const float*)P[41];
    k_out3<<<ceil_div(N0 * 3, 256), 256, 0, stream>>>(h, oW, ob, pbuf[u], N0);
  }

  // --- phi assembly (the 4 index sets partition [0, N0)) ---
  const int q = N0 / 4;                 // 10240
  const int ntri = N0 - 3 * q;          // 10242
  k_phi<<<ceil_div(q, 256), 256, 0, stream>>>(0, d02, q, pbuf[0], pbuf[1], pbuf[2],
                                              r01, r02, r12, phib);
  k_phi<<<ceil_div(q, 256), 256, 0, stream>>>(1, d12, q, pbuf[0], pbuf[1], pbuf[2],
                                              r01, r02, r12, phib);
  k_phi<<<ceil_div(q, 256), 256, 0, stream>>>(2, d01, q, pbuf[0], pbuf[1], pbuf[2],
                                              r01, r02, r12, phib);
  k_phi<<<ceil_div(ntri, 256), 256, 0, stream>>>(3, tri, ntri, pbuf[0], pbuf[1], pbuf[2],
                                                 r01, r02, r12, phib);

  // --- final fusion: writes all 7 outputs ---
  k_final<<<ceil_div(N0, 256), 256, 0, stream>>>(pbuf[0], pbuf[1], pbuf[2], phib, mxyz,
                                                 r01, r02, r12, r20, (float*)d_out, N0);
}